// PerClassAttentionMLPHead_10041633538404
// MI455X (gfx1250) — compile-verified
//
#include <hip/hip_runtime.h>
#include <hip/hip_bf16.h>
#include <math.h>
#include <stdint.h>

// Problem constants (match reference).
#define BB 512
#define SS 512
#define DD 1024
#define CC 33
#define CP 48   // C padded to 3 tiles of 16

typedef float v2f __attribute__((ext_vector_type(2)));
typedef float v8f __attribute__((ext_vector_type(8)));
typedef int v4i __attribute__((__vector_size__(16)));
typedef unsigned int u32;
typedef unsigned long long u64;

#define AS1 __attribute__((address_space(1)))
#define AS3 __attribute__((address_space(3)))

static __device__ __forceinline__ v8f wmma4(v2f a, v2f b, v8f c) {
  // D = A(16x4 f32) * B(4x16 f32) + C(16x16 f32)
  return __builtin_amdgcn_wmma_f32_16x16x4_f32(false, a, false, b, (short)0, c,
                                               false, false);
}

static __device__ __forceinline__ v8f zero8() {
  v8f z;
#pragma unroll
  for (int i = 0; i < 8; ++i) z[i] = 0.f;
  return z;
}

// 16B global -> LDS async copy (ASYNCcnt-tracked DMA, no VGPR data movement).
static __device__ __forceinline__ void async_cp16(void* lds, const void* g) {
#if __has_builtin(__builtin_amdgcn_global_load_async_to_lds_b128)
  __builtin_amdgcn_global_load_async_to_lds_b128(
      (AS1 v4i*)(u64)(uintptr_t)g, (AS3 v4i*)(u32)(uintptr_t)lds, 0, 0);
#else
  asm volatile("global_load_async_to_lds_b128 %0, %1, off"
               :
               : "v"((u32)(uintptr_t)lds), "v"((u64)(uintptr_t)g)
               : "memory");
#endif
}

static __device__ __forceinline__ void wait_async0() {
#if __has_builtin(__builtin_amdgcn_s_wait_asynccnt)
  __builtin_amdgcn_s_wait_asynccnt(0);
#else
  asm volatile("s_wait_asynccnt 0x0" ::: "memory");
#endif
}

// ---------------------------------------------------------------------------
// Kernel 1: scores = q @ x^T * scale  -> scores buffer (B,C,S).
// Grid (B, S/128), block 256 (8 waves).  Wave: 1 s-tile x 3 c-tiles.
// Double-buffered async staging: ql[2][48*36] | xl[2][128*36]
// ---------------------------------------------------------------------------
__global__ void __launch_bounds__(256)
attn_scores(const float* __restrict__ x, const float* __restrict__ q,
            float* __restrict__ sc_g) {
  extern __shared__ float sm[];
  const int QSZ = CP * 36, XSZ = 128 * 36, BUF = QSZ + XSZ;
  float* const ql0 = sm;
  float* const xl0 = sm + QSZ;
  float* const ql1 = sm + BUF;
  float* const xl1 = sm + BUF + QSZ;

  const int b = blockIdx.x;
  const int s0 = blockIdx.y * 128;
  const int t = threadIdx.x;
  const int lane = t & 31, wave = t >> 5;
  const int half = lane >> 4, l16 = lane & 15;

  const float* xb = x + (size_t)b * SS * DD + (size_t)s0 * DD;

  // zero the padded class rows (c >= CC) once; never async-written.
  for (int i = CC * 36 + t; i < QSZ; i += 256) { ql0[i] = 0.f; ql1[i] = 0.f; }

  auto stage = [&](int bsel, int k0) {
    float* qd = bsel ? ql1 : ql0;
    float* xd = bsel ? xl1 : xl0;
    if (t < CC * 8) {
      int c = t >> 3, seg = t & 7;
      async_cp16(qd + c * 36 + seg * 4, q + (size_t)c * DD + k0 + seg * 4);
    }
#pragma unroll
    for (int j = 0; j < 4; ++j) {
      int idx = t + 256 * j;
      int s = idx >> 3, seg = idx & 7;
      async_cp16(xd + s * 36 + seg * 4, xb + (size_t)s * DD + k0 + seg * 4);
    }
  };

  v8f acc[3];
#pragma unroll
  for (int ct = 0; ct < 3; ++ct) acc[ct] = zero8();

  stage(0, 0);
  wait_async0();
  __syncthreads();

  int buf = 0;
  for (int k0 = 0; k0 < DD; k0 += 32) {
    if (k0 + 32 < DD) stage(buf ^ 1, k0 + 32);
    const float* qc = buf ? ql1 : ql0;
    const float* xc = buf ? xl1 : xl0;
#pragma unroll
    for (int kk = 0; kk < 32; kk += 4) {
      v2f bfr;
      bfr.x = xc[(wave * 16 + l16) * 36 + kk + half];
      bfr.y = xc[(wave * 16 + l16) * 36 + kk + 2 + half];
#pragma unroll
      for (int ct = 0; ct < 3; ++ct) {
        v2f afr;
        afr.x = qc[(ct * 16 + l16) * 36 + kk + 2 * half];
        afr.y = qc[(ct * 16 + l16) * 36 + kk + 2 * half + 1];
        acc[ct] = wmma4(afr, bfr, acc[ct]);
      }
    }
    wait_async0();
    __syncthreads();
    buf ^= 1;
  }

  const float scale = 0.03125f;  // 1024^-0.5
  const int scol = s0 + wave * 16 + l16;
#pragma unroll
  for (int ct = 0; ct < 3; ++ct)
#pragma unroll
    for (int i = 0; i < 8; ++i) {
      int c = ct * 16 + i + 8 * half;
      if (c < CC) sc_g[((size_t)b * CC + c) * SS + scol] = acc[ct][i] * scale;
    }
}

// ---------------------------------------------------------------------------
// Kernel 2: softmax over S, in place.  Grid B*C, block 256.
// ---------------------------------------------------------------------------
__global__ void __launch_bounds__(256)
softmax_k(float* __restrict__ sc_g) {
  float* p = sc_g + (size_t)blockIdx.x * SS;
  const int t = threadIdx.x;
  const int lane = t & 31, wave = t >> 5;

  float a0 = p[t], a1 = p[t + 256];
  float m = fmaxf(a0, a1);
#pragma unroll
  for (int msk = 16; msk; msk >>= 1) m = fmaxf(m, __shfl_xor(m, msk, 32));
  __shared__ float rm[8], rsum[8];
  if (lane == 0) rm[wave] = m;
  __syncthreads();
  float M = rm[0];
#pragma unroll
  for (int w = 1; w < 8; ++w) M = fmaxf(M, rm[w]);

  float e0 = __expf(a0 - M), e1 = __expf(a1 - M);
  float s = e0 + e1;
#pragma unroll
  for (int msk = 16; msk; msk >>= 1) s += __shfl_xor(s, msk, 32);
  if (lane == 0) rsum[wave] = s;
  __syncthreads();
  float T = 0.f;
#pragma unroll
  for (int w = 0; w < 8; ++w) T += rsum[w];
  float inv = 1.f / T;
  p[t] = e0 * inv;
  p[t + 256] = e1 * inv;
}

// ---------------------------------------------------------------------------
// Kernel 3: attended = weights(C x S) @ x(S x D).  Grid (B, D/128).
// Double-buffered async staging: wl[2][48*36] | xl[2][32*132]
// ---------------------------------------------------------------------------
__global__ void __launch_bounds__(256)
attn_apply(const float* __restrict__ x, const float* __restrict__ wts,
           float* __restrict__ att) {
  extern __shared__ float sm[];
  const int WSZ = CP * 36, XSZ = 32 * 132, BUF = WSZ + XSZ;
  float* const wl0 = sm;
  float* const xl0 = sm + WSZ;
  float* const wl1 = sm + BUF;
  float* const xl1 = sm + BUF + WSZ;

  const int b = blockIdx.x;
  const int d0 = blockIdx.y * 128;
  const int t = threadIdx.x;
  const int lane = t & 31, wave = t >> 5;
  const int half = lane >> 4, l16 = lane & 15;

  const float* wrow = wts + (size_t)b * CC * SS;
  const float* xb = x + (size_t)b * SS * DD;

  for (int i = CC * 36 + t; i < WSZ; i += 256) { wl0[i] = 0.f; wl1[i] = 0.f; }

  auto stage = [&](int bsel, int s0) {
    float* wd = bsel ? wl1 : wl0;
    float* xd = bsel ? xl1 : xl0;
    if (t < CC * 8) {
      int c = t >> 3, seg = t & 7;
      async_cp16(wd + c * 36 + seg * 4, wrow + (size_t)c * SS + s0 + seg * 4);
    }
#pragma unroll
    for (int j = 0; j < 4; ++j) {
      int idx = t + 256 * j;
      int r = idx >> 5, seg = idx & 31;
      async_cp16(xd + r * 132 + seg * 4,
                 xb + (size_t)(s0 + r) * DD + d0 + seg * 4);
    }
  };

  v8f acc[3];
#pragma unroll
  for (int ct = 0; ct < 3; ++ct) acc[ct] = zero8();

  stage(0, 0);
  wait_async0();
  __syncthreads();

  int buf = 0;
  for (int s0 = 0; s0 < SS; s0 += 32) {
    if (s0 + 32 < SS) stage(buf ^ 1, s0 + 32);
    const float* wc = buf ? wl1 : wl0;
    const float* xc = buf ? xl1 : xl0;
#pragma unroll
    for (int kk = 0; kk < 32; kk += 4) {
      v2f bfr;
      bfr.x = xc[(kk + half) * 132 + wave * 16 + l16];
      bfr.y = xc[(kk + 2 + half) * 132 + wave * 16 + l16];
#pragma unroll
      for (int ct = 0; ct < 3; ++ct) {
        v2f afr;
        afr.x = wc[(ct * 16 + l16) * 36 + kk + 2 * half];
        afr.y = wc[(ct * 16 + l16) * 36 + kk + 2 * half + 1];
        acc[ct] = wmma4(afr, bfr, acc[ct]);
      }
    }
    wait_async0();
    __syncthreads();
    buf ^= 1;
  }

  const int col = d0 + wave * 16 + l16;
#pragma unroll
  for (int ct = 0; ct < 3; ++ct)
#pragma unroll
    for (int i = 0; i < 8; ++i) {
      int c = ct * 16 + i + 8 * half;
      if (c < CC) att[((size_t)b * CC + c) * DD + col] = acc[ct][i];
    }
}

// ---------------------------------------------------------------------------
// Kernel 4: per-(b,c) LayerNorm over D, in place.  Grid B*C, block 256.
// ---------------------------------------------------------------------------
__global__ void __launch_bounds__(256)
layernorm_k(float* __restrict__ att, const float* __restrict__ gamma,
            const float* __restrict__ beta) {
  const int bc = blockIdx.x;
  const int c = bc % CC;
  float* row = att + (size_t)bc * DD;
  const int t = threadIdx.x;

  float v[4], s = 0.f, s2 = 0.f;
#pragma unroll
  for (int i = 0; i < 4; ++i) {
    v[i] = row[t + 256 * i];
    s += v[i];
    s2 += v[i] * v[i];
  }
#pragma unroll
  for (int m = 16; m; m >>= 1) {
    s += __shfl_xor(s, m, 32);
    s2 += __shfl_xor(s2, m, 32);
  }
  __shared__ float rs[8], rs2[8];
  if ((t & 31) == 0) { rs[t >> 5] = s; rs2[t >> 5] = s2; }
  __syncthreads();
  float ts = 0.f, ts2 = 0.f;
#pragma unroll
  for (int w = 0; w < 8; ++w) { ts += rs[w]; ts2 += rs2[w]; }
  const float mu = ts * (1.f / DD);
  const float var = ts2 * (1.f / DD) - mu * mu;
  const float rstd = rsqrtf(var + 1e-5f);
#pragma unroll
  for (int i = 0; i < 4; ++i) {
    int d = t + 256 * i;
    row[d] = (v[i] - mu) * rstd * gamma[(size_t)c * DD + d] +
             beta[(size_t)c * DD + d];
  }
}

// ---------------------------------------------------------------------------
// Kernel 5: per-class MLP + logits.  Grid (C, B/64), block 256 (8 waves).
// Double-buffered async staging: hl[2][64*36] | wl[2][32*68] | wsum[8*16]
// Fused bias + exact GELU + W2 dot, deterministic reduction.
// ---------------------------------------------------------------------------
__global__ void __launch_bounds__(256)
mlp_head(const float* __restrict__ hn, const float* __restrict__ W1,
         const float* __restrict__ b1, const float* __restrict__ W2,
         const float* __restrict__ b2, float* __restrict__ out) {
  extern __shared__ float sm[];
  const int HSZ = 64 * 36, WSZ = 32 * 68, BUF = HSZ + WSZ;
  float* const hl0 = sm;
  float* const wl0 = sm + HSZ;
  float* const hl1 = sm + BUF;
  float* const wl1 = sm + BUF + HSZ;
  float* const wsum = sm + 2 * BUF;  // [8][16]

  const int c = blockIdx.x;
  const int b0 = blockIdx.y * 64;
  const int t = threadIdx.x;
  const int lane = t & 31, wave = t >> 5;
  const int half = lane >> 4, l16 = lane & 15;
  const int tr = wave >> 1;        // tile row (16 batch rows)
  const int tc0 = (wave & 1) * 2;  // first of 2 col tiles

  auto stage = [&](int bsel, int e0, int k0) {
    float* hd = bsel ? hl1 : hl0;
    float* wd = bsel ? wl1 : wl0;
#pragma unroll
    for (int j = 0; j < 2; ++j) {
      int idx = t + 256 * j;
      int m = idx >> 3, seg = idx & 7;
      async_cp16(hd + m * 36 + seg * 4,
                 hn + ((size_t)(b0 + m) * CC + c) * DD + k0 + seg * 4);
    }
#pragma unroll
    for (int j = 0; j < 2; ++j) {
      int idx = t + 256 * j;
      int r = idx >> 4, seg = idx & 15;
      async_cp16(wd + r * 68 + seg * 4,
                 W1 + ((size_t)c * DD + k0 + r) * DD + e0 + seg * 4);
    }
  };

  float plog[8];
#pragma unroll
  for (int i = 0; i < 8; ++i) plog[i] = 0.f;

  stage(0, 0, 0);
  wait_async0();
  __syncthreads();

  int buf = 0;
  for (int e0 = 0; e0 < DD; e0 += 64) {
    v8f acc[2];
    acc[0] = zero8();
    acc[1] = zero8();
    for (int k0 = 0; k0 < DD; k0 += 32) {
      int ne = e0, nk = k0 + 32;
      if (nk == DD) { ne = e0 + 64; nk = 0; }
      if (ne < DD) stage(buf ^ 1, ne, nk);
      const float* hc = buf ? hl1 : hl0;
      const float* wc = buf ? wl1 : wl0;
#pragma unroll
      for (int kk = 0; kk < 32; kk += 4) {
        v2f a;
        a.x = hc[(tr * 16 + l16) * 36 + kk + 2 * half];
        a.y = hc[(tr * 16 + l16) * 36 + kk + 2 * half + 1];
#pragma unroll
        for (int nt = 0; nt < 2; ++nt) {
          int col = (tc0 + nt) * 16 + l16;
          v2f bb;
          bb.x = wc[(kk + half) * 68 + col];
          bb.y = wc[(kk + 2 + half) * 68 + col];
          acc[nt] = wmma4(a, bb, acc[nt]);
        }
      }
      wait_async0();
      __syncthreads();
      buf ^= 1;
    }
    // fused epilogue: bias + exact GELU + W2 dot, accumulate per-row partials
#pragma unroll
    for (int nt = 0; nt < 2; ++nt) {
      int e = e0 + (tc0 + nt) * 16 + l16;
      float b1v = b1[(size_t)c * DD + e];
      float w2v = W2[(size_t)c * DD + e];
#pragma unroll
      for (int i = 0; i < 8; ++i) {
        float h = acc[nt][i] + b1v;
        float g = 0.5f * h * (1.f + erff(h * 0.70710678118f));
        plog[i] += g * w2v;
      }
    }
  }

  // reduce across the 16 lanes sharing a row group (deterministic)
#pragma unroll
  for (int msk = 1; msk < 16; msk <<= 1)
#pragma unroll
    for (int i = 0; i < 8; ++i) plog[i] += __shfl_xor(plog[i], msk, 32);

  if (l16 == 0) {
#pragma unroll
    for (int i = 0; i < 8; ++i) wsum[wave * 16 + i + 8 * half] = plog[i];
  }
  __syncthreads();

  if (t < 64) {
    int m = t, trr = m >> 4, r = m & 15;
    float v = wsum[(2 * trr) * 16 + r] + wsum[(2 * trr + 1) * 16 + r] + b2[c];
    out[(size_t)(b0 + m) * CC + c] = v;
  }
}

// ---------------------------------------------------------------------------
extern "C" void kernel_launch(void* const* d_in, const int* in_sizes, int n_in,
                              void* d_out, int out_size, void* d_ws,
                              size_t ws_size, hipStream_t stream) {
  const float* x     = (const float*)d_in[0];
  const float* q     = (const float*)d_in[1];
  const float* gamma = (const float*)d_in[2];
  const float* beta  = (const float*)d_in[3];
  const float* W1    = (const float*)d_in[4];
  const float* b1    = (const float*)d_in[5];
  const float* W2    = (const float*)d_in[6];
  const float* b2    = (const float*)d_in[7];
  float* out = (float*)d_out;

  float* wts = (float*)d_ws;                 // (B,C,S)  scores -> weights
  float* att = wts + (size_t)BB * CC * SS;   // (B,C,D)

  const size_t sm1 = (size_t)(2 * (CP * 36 + 128 * 36)) * sizeof(float);
  const size_t sm3 = (size_t)(2 * (CP * 36 + 32 * 132)) * sizeof(float);
  const size_t sm5 = (size_t)(2 * (64 * 36 + 32 * 68) + 8 * 16) * sizeof(float);

  attn_scores<<<dim3(BB, SS / 128), 256, sm1, stream>>>(x, q, wts);
  softmax_k<<<BB * CC, 256, 0, stream>>>(wts);
  attn_apply<<<dim3(BB, DD / 128), 256, sm3, stream>>>(x, wts, att);
  layernorm_k<<<BB * CC, 256, 0, stream>>>(att, gamma, beta);
  mlp_head<<<dim3(CC, BB / 64), 256, sm5, stream>>>(att, W1, b1, W2, b2, out);
}